// LSTMMod_68676527063622
// MI455X (gfx1250) — compile-verified
//
#include <hip/hip_runtime.h>
#include <math.h>

#define T_LEN 32768
#define CLS   1024
#define KS    32            // K-splits for the GEMV (parallelism to saturate HBM)
#define CG    (CLS / 16)    // class groups of 16 (WMMA N/M tile)

typedef __attribute__((ext_vector_type(2))) float v2f;
typedef __attribute__((ext_vector_type(8))) float v8f;

__device__ __forceinline__ float sigmoid_fast(float x) {
    return 1.0f / (1.0f + __expf(-x));
}

// ---------------------------------------------------------------------------
// Kernel 1: 3-layer LSTM scan, software-pipelined across layers.
// Single thread; layer l at iteration n processes time (n - l), so the three
// layer updates inside one iteration are mutually independent -> 3x ILP on
// the serial critical path (which is the whole problem's bottleneck).
// ---------------------------------------------------------------------------
__global__ void lstm_scan_k(const float* __restrict__ x,  const float* __restrict__ h0v,
                            const float* __restrict__ c0v, const float* __restrict__ w_ih,
                            const float* __restrict__ w_hh, const float* __restrict__ b_ih,
                            const float* __restrict__ b_hh, float* __restrict__ outs) {
    float wi[12], wh[12], bs[12];
#pragma unroll
    for (int i = 0; i < 12; ++i) {
        wi[i] = w_ih[i];
        wh[i] = w_hh[i];
        bs[i] = b_ih[i] + b_hh[i];
    }
    float h[3] = {h0v[0], h0v[1], h0v[2]};
    float c[3] = {c0v[0], c0v[1], c0v[2]};

    auto step = [&](int l, float in) -> float {
        const float* wl = wi + 4 * l;
        const float* hl = wh + 4 * l;
        const float* bl = bs + 4 * l;
        float g0 = fmaf(wl[0], in, fmaf(hl[0], h[l], bl[0]));
        float g1 = fmaf(wl[1], in, fmaf(hl[1], h[l], bl[1]));
        float g2 = fmaf(wl[2], in, fmaf(hl[2], h[l], bl[2]));
        float g3 = fmaf(wl[3], in, fmaf(hl[3], h[l], bl[3]));
        float ig = sigmoid_fast(g0);
        float fg = sigmoid_fast(g1);
        float gg = tanhf(g2);
        float og = sigmoid_fast(g3);
        c[l] = fmaf(fg, c[l], ig * gg);
        h[l] = og * tanhf(c[l]);
        return h[l];
    };

    // Prologue: fill the 2-deep pipeline.
    float p01 = step(0, x[0]);                 // Y0[0]
    float y0  = step(0, x[1]);                 // Y0[1]
    float p12 = step(1, p01);                  // Y1[0]
    p01 = y0;

    // Steady state: three independent chains per iteration.
    for (int n = 2; n < T_LEN; ++n) {
        float a0 = step(0, x[n]);              // Y0[n]
        float a1 = step(1, p01);               // Y1[n-1]
        float a2 = step(2, p12);               // Y2[n-2]
        outs[n - 2] = a2;
        p01 = a0;
        p12 = a1;
    }

    // Epilogue: drain.
    float a1 = step(1, p01);                   // Y1[T-1]
    float a2 = step(2, p12);                   // Y2[T-2]
    outs[T_LEN - 2] = a2;
    a2 = step(2, a1);                          // Y2[T-1]
    outs[T_LEN - 1] = a2;
}

// ---------------------------------------------------------------------------
// Kernel 2: logits[c] = sum_t outs[t] * W[c,t] via V_WMMA_F32_16X16X4_F32.
// A = W tile (16 classes x 4 timesteps), B = outs chunk broadcast across N.
// Per ISA A-layout: VGPR0 holds K=0 (lanes 0-15) / K=2 (lanes 16-31);
// VGPR1 holds K=1 / K=3 -> each lane does one b64 load of 2 consecutive t.
// D column N=0: lane 0 carries M=0..7 (VGPR 0..7), lane 16 carries M=8..15.
// K is split KS ways across blockIdx.y; partials merged by f32 atomics.
// ---------------------------------------------------------------------------
__global__ void gemv_wmma_k(const float* __restrict__ W, const float* __restrict__ outs,
                            float* __restrict__ logits) {
    const int lane   = threadIdx.x;            // 32-thread (1-wave) blocks
    const int c0     = blockIdx.x * 16;
    const int kchunk = T_LEN / KS;             // 1024
    const int kbase  = blockIdx.y * kchunk;
    const int row    = c0 + (lane & 15);
    const int koff   = (lane >> 4) << 1;       // 0 for lanes 0-15, 2 for 16-31

    const float* wrow = W + (size_t)row * T_LEN;

    v8f acc0 = {0.f, 0.f, 0.f, 0.f, 0.f, 0.f, 0.f, 0.f};
    v8f acc1 = {0.f, 0.f, 0.f, 0.f, 0.f, 0.f, 0.f, 0.f};

#pragma unroll 4
    for (int t = kbase; t < kbase + kchunk; t += 8) {
        int t0 = t + koff;
        int t1 = t + 4 + koff;
        v2f a0 = *(const v2f*)(wrow + t0);
        v2f b0 = *(const v2f*)(outs + t0);
        v2f a1 = *(const v2f*)(wrow + t1);
        v2f b1 = *(const v2f*)(outs + t1);
        acc0 = __builtin_amdgcn_wmma_f32_16x16x4_f32(false, a0, false, b0,
                                                     (short)0, acc0, false, false);
        acc1 = __builtin_amdgcn_wmma_f32_16x16x4_f32(false, a1, false, b1,
                                                     (short)0, acc1, false, false);
    }

    if ((lane & 15) == 0) {                    // lanes 0 and 16 hold column N=0
        float* dst = logits + c0 + (lane >> 4) * 8;
#pragma unroll
        for (int j = 0; j < 8; ++j) {
            atomicAdd(dst + j, acc0[j] + acc1[j]);
        }
    }
}

// ---------------------------------------------------------------------------
// Kernel 3: out = softmax(logits + bias) over CLS=1024.
// ---------------------------------------------------------------------------
__global__ void softmax_k(const float* __restrict__ logits, const float* __restrict__ bias,
                          float* __restrict__ out) {
    __shared__ float red[CLS];
    const int i = threadIdx.x;
    float v = logits[i] + bias[i];
    red[i] = v;
    __syncthreads();
    for (int s = CLS / 2; s > 0; s >>= 1) {
        if (i < s) red[i] = fmaxf(red[i], red[i + s]);
        __syncthreads();
    }
    float m = red[0];
    __syncthreads();
    float e = __expf(v - m);
    red[i] = e;
    __syncthreads();
    for (int s = CLS / 2; s > 0; s >>= 1) {
        if (i < s) red[i] += red[i + s];
        __syncthreads();
    }
    float inv = 1.0f / red[0];
    out[i] = e * inv;
}

__global__ void zero_k(float* __restrict__ p) {
    p[threadIdx.x] = 0.0f;
}

// ---------------------------------------------------------------------------
extern "C" void kernel_launch(void* const* d_in, const int* in_sizes, int n_in,
                              void* d_out, int out_size, void* d_ws, size_t ws_size,
                              hipStream_t stream) {
    (void)in_sizes; (void)n_in; (void)out_size; (void)ws_size;

    const float* x     = (const float*)d_in[0];
    const float* h0    = (const float*)d_in[1];
    const float* c0    = (const float*)d_in[2];
    const float* w_ih  = (const float*)d_in[3];
    const float* w_hh  = (const float*)d_in[4];
    const float* b_ih  = (const float*)d_in[5];
    const float* b_hh  = (const float*)d_in[6];
    const float* lin_w = (const float*)d_in[7];
    const float* lin_b = (const float*)d_in[8];
    float*       outp  = (float*)d_out;

    float* outs   = (float*)d_ws;              // T_LEN floats
    float* logits = outs + T_LEN;              // CLS floats (16B aligned)

    hipLaunchKernelGGL(zero_k, dim3(1), dim3(CLS), 0, stream, logits);
    hipLaunchKernelGGL(lstm_scan_k, dim3(1), dim3(1), 0, stream,
                       x, h0, c0, w_ih, w_hh, b_ih, b_hh, outs);
    hipLaunchKernelGGL(gemv_wmma_k, dim3(CG, KS), dim3(32), 0, stream,
                       lin_w, outs, logits);
    hipLaunchKernelGGL(softmax_k, dim3(1), dim3(CLS), 0, stream,
                       logits, lin_b, outp);
}